// MeshMask_8022998909310
// MI455X (gfx1250) — compile-verified
//
#include <hip/hip_runtime.h>
#include <hip/hip_bf16.h>

#define KSAMP     12
#define BIG_F     1e10f
#define EPSZ      0.001f
#define TRI_BLK   256                      // triangles per block == blockDim.x
#define TRI_WORDS (TRI_BLK * 3)            // 768 int32 = 3072 B staged per block

typedef __attribute__((ext_vector_type(4))) unsigned int uint4v;
typedef __attribute__((ext_vector_type(4))) int          int4v;
typedef __attribute__((ext_vector_type(8))) int          int8v;

// ---------------------------------------------------------------------------
// Kernel 1: initialize z-buffer to BIG. H*W read on-device (graph-capture
// safe), fixed grid + stride loop -> deterministic launch shape.
// ---------------------------------------------------------------------------
__global__ __launch_bounds__(256) void zb_fill(float* __restrict__ zbuf,
                                               const int* __restrict__ Hp,
                                               const int* __restrict__ Wp) {
    const int total  = Hp[0] * Wp[0];
    const int stride = gridDim.x * blockDim.x;
    for (int i = blockIdx.x * blockDim.x + threadIdx.x; i < total; i += stride)
        zbuf[i] = BIG_F;
}

// ---------------------------------------------------------------------------
// Kernel 2: rasterize. One thread per triangle; each block stages its 256
// triangles' index data (3072 B, perfectly linear) into LDS with a single
// Tensor-Data-Mover 1-D tile load issued by thread 0 (TDM ignores EXEC, so
// it sits behind a branch other waves never reach), then s_wait_tensorcnt 0
// + workgroup barrier. Threads read indices from LDS, gather vertices from
// the L2-resident vertex array, and scatter min-depth via
// global_atomic_min_u32 (uint ordering == float ordering after -0 clamp).
// ---------------------------------------------------------------------------
__global__ __launch_bounds__(TRI_BLK) void zb_raster(const float* __restrict__ verts,
                                                     const int*   __restrict__ tris,
                                                     int T,
                                                     const int* __restrict__ Hp,
                                                     const int* __restrict__ Wp,
                                                     unsigned int* __restrict__ zbuf) {
    __shared__ int smemTri[TRI_WORDS];

    const int lt   = threadIdx.x;                 // triangle within block
    const int tri0 = blockIdx.x * TRI_BLK;        // first triangle of block
    const int t    = tri0 + lt;

    // ---- TDM stage: 1-D tile of 768 dwords tris[3*tri0 ..] -> LDS ---------
    if (lt == 0) {
        const unsigned long long ga = (unsigned long long)(const void*)(tris + 3 * tri0);
        const unsigned lds_off = (unsigned)(unsigned long long)(const void*)&smemTri[0];

        long long remain = (long long)3 * T - (long long)3 * tri0;  // >0 here
        unsigned td0 = (remain >= TRI_WORDS) ? (unsigned)TRI_WORDS : (unsigned)remain;

        // D# group 0: count=1 | lds_addr | global_addr[56:0] | type=2
        uint4v g0 = { 1u,
                      lds_off,
                      (unsigned)(ga & 0xffffffffu),
                      (unsigned)((ga >> 32) & 0x01ffffffu) | (2u << 30) };

        // D# group 1: data_size=4B(code 2); tensor_dim0=td0; tensor_dim1=1;
        // tile_dim0=768; tile_dim1/2=0 (1-D); dim0_stride=768; rest 0.
        int8v g1 = { (int)(2u << 16),                                   // [17:16] data_size
                     (int)((td0 & 0xffffu) << 16),                      // [63:48] dim0 lo
                     (int)(((td0 >> 16) & 0xffffu) | (1u << 16)),       // [79:64] dim0 hi | [95:80] dim1 lo
                     (int)(((unsigned)TRI_WORDS) << 16),                // [111:96] dim1 hi=0 | [127:112] tile_dim0
                     0,                                                 // tile_dim1=0, tile_dim2=0
                     (int)TRI_WORDS,                                    // dim0_stride lo32
                     0, 0 };                                            // stride hi, dim1_stride

        int4v gz = { 0, 0, 0, 0 };                                      // groups 2/3 unused

#if defined(__clang_major__) && (__clang_major__ >= 23)
        int8v gz8 = { 0, 0, 0, 0, 0, 0, 0, 0 };
        __builtin_amdgcn_tensor_load_to_lds(g0, g1, gz, gz, gz8, 0);
#else
        __builtin_amdgcn_tensor_load_to_lds(g0, g1, gz, gz, 0);
#endif
        __builtin_amdgcn_s_wait_tensorcnt(0);     // tile landed in LDS
    }
    __syncthreads();

    if (t >= T) return;

    const int   W   = Wp[0];
    const float Wm1 = (float)(W - 1);
    const float Hm1 = (float)(Hp[0] - 1);

    const int i0 = smemTri[3 * lt + 0];
    const int i1 = smemTri[3 * lt + 1];
    const int i2 = smemTri[3 * lt + 2];

    const float ax = verts[3 * i0 + 0], ay = verts[3 * i0 + 1], az = verts[3 * i0 + 2];
    const float bx = verts[3 * i1 + 0], by = verts[3 * i1 + 1], bz = verts[3 * i1 + 2];
    const float cx = verts[3 * i2 + 0], cy = verts[3 * i2 + 1], cz = verts[3 * i2 + 2];

    // area = (v2x-v0x)*(v1y-v0y) - (v2y-v0y)*(v1x-v0x); degenerate -> no valid samples
    const float area = (cx - ax) * (by - ay) - (cy - ay) * (bx - ax);
    if (!(fabsf(area) > 1e-8f)) return;

    const float xmin = fminf(ax, fminf(bx, cx)), xmax = fmaxf(ax, fmaxf(bx, cx));
    const float ymin = fminf(ay, fminf(by, cy)), ymax = fmaxf(ay, fmaxf(by, cy));
    const float dxr = xmax - xmin, dyr = ymax - ymin;
    const float inv = 1.0f / (float)(KSAMP - 1);           // linspace(0,1,K) step

    for (int i = 0; i < KSAMP; ++i) {
        const float px = xmin + ((float)i * inv) * dxr;
        if (!((px >= 0.0f) && (px <= Wm1))) continue;       // column off-image in x
        const int ix = (int)fminf(fmaxf(rintf(px), 0.0f), Wm1);  // round-half-even

        for (int j = 0; j < KSAMP; ++j) {
            const float py = ymin + ((float)j * inv) * dyr;
            if (!((py >= 0.0f) && (py <= Hm1))) continue;

            // edge(v1,v2), edge(v2,v0), edge(v0,v1) — exact reference formulas
            const float w0 = (px - bx) * (cy - by) - (py - by) * (cx - bx);
            const float w1 = (px - cx) * (ay - cy) - (py - cy) * (ax - cx);
            const float w2 = (px - ax) * (by - ay) - (py - ay) * (bx - ax);

            const bool inside = ((w0 >= 0.0f) && (w1 >= 0.0f) && (w2 >= 0.0f)) ||
                                ((w0 <= 0.0f) && (w1 <= 0.0f) && (w2 <= 0.0f));
            if (!inside) continue;

            float depth = (w0 * az + w1 * bz + w2 * cz) / area;
            depth = fmaxf(depth, 0.0f);   // kill -0.0 so uint-min == float-min

            const int iy = (int)fminf(fmaxf(rintf(py), 0.0f), Hm1);
            atomicMin(&zbuf[iy * W + ix], __float_as_uint(depth));
        }
    }
}

// ---------------------------------------------------------------------------
// Kernel 3: per-vertex visibility test against the z-buffer.
// ---------------------------------------------------------------------------
__global__ __launch_bounds__(256) void vis_mask(const float* __restrict__ verts,
                                                int N,
                                                const int* __restrict__ Hp,
                                                const int* __restrict__ Wp,
                                                const float* __restrict__ zbuf,
                                                float* __restrict__ out) {
    const int n = blockIdx.x * blockDim.x + threadIdx.x;
    if (n >= N) return;

    const int   W   = Wp[0];
    const float Wm1 = (float)(W - 1);
    const float Hm1 = (float)(Hp[0] - 1);

    const float x = verts[3 * n + 0];
    const float y = verts[3 * n + 1];
    const float z = verts[3 * n + 2];

    const bool in_img = (x >= 0.0f) && (x <= Wm1) && (y >= 0.0f) && (y <= Hm1);
    const int ix = (int)fminf(fmaxf(rintf(x), 0.0f), Wm1);
    const int iy = (int)fminf(fmaxf(rintf(y), 0.0f), Hm1);

    const float zb = zbuf[iy * W + ix];   // indices clipped -> always safe
    out[n] = (in_img && (z <= zb + EPSZ)) ? 1.0f : 0.0f;
}

// ---------------------------------------------------------------------------
// Launch: fill -> raster -> visibility, stream-ordered (graph-capture safe).
// zbuf lives in workspace (H*W floats = 1 MB for 512x512, L2-resident).
// ---------------------------------------------------------------------------
extern "C" void kernel_launch(void* const* d_in, const int* in_sizes, int n_in,
                              void* d_out, int out_size, void* d_ws, size_t ws_size,
                              hipStream_t stream) {
    const float* verts = (const float*)d_in[0];
    const int*   tris  = (const int*)d_in[1];
    const int*   Hp    = (const int*)d_in[2];
    const int*   Wp    = (const int*)d_in[3];

    const int N = in_sizes[0] / 3;   // vertices
    const int T = in_sizes[1] / 3;   // triangles

    float* zbuf = (float*)d_ws;

    zb_fill<<<2048, 256, 0, stream>>>(zbuf, Hp, Wp);

    const int rblocks = (T + TRI_BLK - 1) / TRI_BLK;
    zb_raster<<<rblocks, TRI_BLK, 0, stream>>>(verts, tris, T, Hp, Wp,
                                               (unsigned int*)zbuf);

    const int vblocks = (N + 255) / 256;
    vis_mask<<<vblocks, 256, 0, stream>>>(verts, N, Hp, Wp, zbuf, (float*)d_out);
}